// Group_28200755265797
// MI455X (gfx1250) — compile-verified
//
#include <hip/hip_runtime.h>
#include <hip/hip_bf16.h>

// Problem constants (fixed by the reference)
#define B_SZ   8
#define N_PTS  8192
#define NGROUP 512     // G
#define KNN_K  32      // M
#define CT     16      // centers per KNN block (one WMMA M-tile)
#define CHUNK  512     // points per KNN chunk
#define DROWS  516     // padded LDS row stride (floats): 8*516 % 64 == 32 -> half-waves hit disjoint banks

typedef __attribute__((ext_vector_type(2))) float v2f;
typedef __attribute__((ext_vector_type(8))) float v8f;

// ---------------------------------------------------------------------------
// Phase 1: farthest point sampling. One 1024-thread block per batch; the whole
// 8192x3 point set is register-resident (8 points/thread). 511 sequential
// rounds of: distance update + block argmax (wave shfl reduce, then LDS).
// ---------------------------------------------------------------------------
__global__ __launch_bounds__(1024)
void fps_kernel(const float* __restrict__ xyz, float* __restrict__ center)
{
    const int b = blockIdx.x;
    const int t = threadIdx.x;
    const int lane = t & 31;
    const int wave = t >> 5;
    const float* pts = xyz + (size_t)b * N_PTS * 3;

    float px[8], py[8], pz[8], dmin[8];
#pragma unroll
    for (int j = 0; j < 8; ++j) {
        const int p = t + 1024 * j;
        px[j] = pts[p * 3 + 0];
        py[j] = pts[p * 3 + 1];
        pz[j] = pts[p * 3 + 2];
        dmin[j] = __builtin_inff();
    }

    __shared__ float s_last[3];
    __shared__ float s_val[32];
    __shared__ int   s_idx[32];
    __shared__ int   s_win;

    float* ctr_b = center + (size_t)b * NGROUP * 3;

    // selected[0] = 0 (deterministic start); thread 0 owns point 0.
    if (t == 0) {
        s_last[0] = px[0]; s_last[1] = py[0]; s_last[2] = pz[0];
        ctr_b[0] = px[0];  ctr_b[1] = py[0];  ctr_b[2] = pz[0];
    }
    __syncthreads();

    for (int i = 1; i < NGROUP; ++i) {
        const float lx = s_last[0], ly = s_last[1], lz = s_last[2];
        float bv = -1.0f; int bi = 0x7fffffff;
#pragma unroll
        for (int j = 0; j < 8; ++j) {
            const float dx = px[j] - lx, dy = py[j] - ly, dz = pz[j] - lz;
            const float dist = dx * dx + dy * dy + dz * dz;
            dmin[j] = fminf(dmin[j], dist);
            const int p = t + 1024 * j;
            if (dmin[j] > bv || (dmin[j] == bv && p < bi)) { bv = dmin[j]; bi = p; }
        }
        // wave32 argmax (prefer larger value; tie -> smaller index, matching argmax-first)
#pragma unroll
        for (int off = 16; off; off >>= 1) {
            const float ov = __shfl_xor(bv, off, 32);
            const int   oi = __shfl_xor(bi, off, 32);
            if (ov > bv || (ov == bv && oi < bi)) { bv = ov; bi = oi; }
        }
        if (lane == 0) { s_val[wave] = bv; s_idx[wave] = bi; }
        __syncthreads();
        if (wave == 0) {
            bv = s_val[lane]; bi = s_idx[lane];
#pragma unroll
            for (int off = 16; off; off >>= 1) {
                const float ov = __shfl_xor(bv, off, 32);
                const int   oi = __shfl_xor(bi, off, 32);
                if (ov > bv || (ov == bv && oi < bi)) { bv = ov; bi = oi; }
            }
            if (lane == 0) s_win = bi;
        }
        __syncthreads();
        const int win = s_win;
        if (t == (win & 1023)) {                 // owner publishes coords
            const int j = win >> 10;
            s_last[0] = px[j]; s_last[1] = py[j]; s_last[2] = pz[j];
            ctr_b[i * 3 + 0] = px[j];
            ctr_b[i * 3 + 1] = py[j];
            ctr_b[i * 3 + 2] = pz[j];
        }
        __syncthreads();
    }
}

// ---------------------------------------------------------------------------
// Phase 2: KNN + gather. Block = 512 threads (16 waves) handles 16 centers of
// one batch. Distances via V_WMMA_F32_16X16X4_F32:
//   A row m = [cx, cy, cz, 1], B col n = [-2px, -2py, -2pz, |p|^2]
//   D[m][n] = |p|^2 - 2 c.p   (add |c|^2 at selection -> full squared distance)
// Fragment construction is branchless (v_cndmask, no EXEC save/restore around
// the WMMA). Results staged in padded LDS; wave w selects top-32 for center w
// with per-lane sorted insertion lists + a 32-round wave argmin merge.
// ---------------------------------------------------------------------------
__global__ __launch_bounds__(512)
void knn_kernel(const float* __restrict__ xyz, const float* __restrict__ center,
                float* __restrict__ nbh)
{
    const int bid  = blockIdx.x;
    const int b    = bid >> 5;        // / (NGROUP/CT)
    const int gt   = bid & 31;        // center-tile within batch
    const int t    = threadIdx.x;
    const int lane = t & 31;
    const int wave = t >> 5;          // 0..15 -> which center this wave selects
    const bool hi  = (lane & 16) != 0; // K half for WMMA A/B fragments
    const int m    = lane & 15;        // row/col within 16-tile

    __shared__ float s_cx[CT], s_cy[CT], s_cz[CT], s_cn[CT];
    __shared__ float s_dist[CT * DROWS];

    const float* pts = xyz + (size_t)b * N_PTS * 3;
    const float* ctr = center + ((size_t)b * NGROUP + (size_t)gt * CT) * 3;

    if (t < CT) {
        const float cx = ctr[t * 3 + 0], cy = ctr[t * 3 + 1], cz = ctr[t * 3 + 2];
        s_cx[t] = cx; s_cy[t] = cy; s_cz[t] = cz;
        s_cn[t] = cx * cx + cy * cy + cz * cz;
    }
    __syncthreads();

    // A fragment: 16x4 f32 -> 2 VGPRs; lanes 0-15 hold K=0,1; lanes 16-31 hold K=2,3
    // Branchless: per-lane selects instead of divergent control flow.
    v2f afrag;
    {
        const float cx = s_cx[m], cy = s_cy[m], cz = s_cz[m];
        afrag.x = hi ? cz : cx;
        afrag.y = hi ? 1.0f : cy;
    }

    const float cnorm = s_cn[wave];

    // per-lane sorted top-32 candidate list
    float kd[KNN_K];
    int   ki[KNN_K];
#pragma unroll
    for (int j = 0; j < KNN_K; ++j) { kd[j] = __builtin_inff(); ki[j] = 0x7fffffff; }

    for (int c = 0; c < N_PTS / CHUNK; ++c) {
        const int base = c * CHUNK;

        // --- WMMA phase: 32 column sub-tiles of 16 points; 2 per wave ---
#pragma unroll
        for (int s = 0; s < 2; ++s) {
            const int tile = wave * 2 + s;
            const int p    = base + tile * 16 + m;
            const float x = pts[p * 3 + 0];
            const float y = pts[p * 3 + 1];
            const float z = pts[p * 3 + 2];
            // Branchless B fragment: lanes 0-15 -> {-2x,-2y}; lanes 16-31 -> {-2z,|p|^2}
            const float pn = x * x + y * y + z * z;
            v2f bfrag;
            bfrag.x = -2.0f * (hi ? z : x);
            bfrag.y = hi ? pn : (-2.0f * y);
            v8f cacc = {};
            v8f d = __builtin_amdgcn_wmma_f32_16x16x4_f32(
                /*neg_a=*/false, afrag, /*neg_b=*/false, bfrag,
                /*c_mod=*/(short)0, cacc, /*reuse_a=*/false, /*reuse_b=*/false);
            // D layout: VGPR v -> M = v + 8*(lane>=16), N = m
            const int col = tile * 16 + m;
            const int rowbase = hi ? 8 : 0;
#pragma unroll
            for (int v = 0; v < 8; ++v) {
                s_dist[(rowbase + v) * DROWS + col] = d[v];
            }
        }
        __syncthreads();

        // --- selection phase: wave w scans row w of the chunk ---
        const float* drow = &s_dist[wave * DROWS];
#pragma unroll 4
        for (int q = 0; q < CHUNK / 32; ++q) {
            const int   col = lane + q * 32;
            const float dd  = drow[col] + cnorm;
            const int   idx = base + col;
            if (dd < kd[KNN_K - 1] || (dd == kd[KNN_K - 1] && idx < ki[KNN_K - 1])) {
                int j = KNN_K - 1;
                while (j > 0 && (kd[j - 1] > dd || (kd[j - 1] == dd && ki[j - 1] > idx))) {
                    kd[j] = kd[j - 1]; ki[j] = ki[j - 1]; --j;
                }
                kd[j] = dd; ki[j] = idx;
            }
        }
        __syncthreads();
    }

    // --- merge: 32 rounds of wave32 lexicographic argmin over lane heads ---
    int h = 0;
    int myIdx = 0;
    for (int r = 0; r < KNN_K; ++r) {
        float cd = (h < KNN_K) ? kd[h] : __builtin_inff();
        int   ci = (h < KNN_K) ? ki[h] : 0x7fffffff;
        int   cl = lane;
#pragma unroll
        for (int off = 16; off; off >>= 1) {
            const float od = __shfl_xor(cd, off, 32);
            const int   oi = __shfl_xor(ci, off, 32);
            const int   ol = __shfl_xor(cl, off, 32);
            if (od < cd || (od == cd && oi < ci)) { cd = od; ci = oi; cl = ol; }
        }
        if (lane == cl) ++h;        // winner lane consumes its head
        if (lane == r)  myIdx = ci; // lane r keeps rank-r neighbor index
    }

    // --- gather + recenter: lane r writes neighbor r of center `wave` ---
    const int g = gt * CT + wave;
    const float ox = pts[myIdx * 3 + 0] - s_cx[wave];
    const float oy = pts[myIdx * 3 + 1] - s_cy[wave];
    const float oz = pts[myIdx * 3 + 2] - s_cz[wave];
    float* o = nbh + (((size_t)b * NGROUP + g) * KNN_K + lane) * 3;
    o[0] = ox; o[1] = oy; o[2] = oz;
}

// ---------------------------------------------------------------------------
extern "C" void kernel_launch(void* const* d_in, const int* in_sizes, int n_in,
                              void* d_out, int out_size, void* d_ws, size_t ws_size,
                              hipStream_t stream)
{
    (void)in_sizes; (void)n_in; (void)out_size; (void)d_ws; (void)ws_size;
    const float* xyz = (const float*)d_in[0];
    float* out = (float*)d_out;
    float* nbh    = out;                                       // [8][512][32][3]
    float* center = out + (size_t)B_SZ * NGROUP * KNN_K * 3;   // [8][512][3]

    fps_kernel<<<B_SZ, 1024, 0, stream>>>(xyz, center);
    knn_kernel<<<B_SZ * (NGROUP / CT), 512, 0, stream>>>(xyz, center, nbh);
}